// GaussianUpdater_4226247819638
// MI455X (gfx1250) — compile-verified
//
#include <hip/hip_runtime.h>
#include <math.h>

#define RES   256
#define FEAT  32
#define NG    500000
#define HW    (RES*RES)          // 65536
#define PLANE_ELEMS (FEAT*HW)    // 2097152

typedef float v2f __attribute__((ext_vector_type(2)));
typedef float v8f __attribute__((ext_vector_type(8)));

// ---------------- workspace layout (floats) ----------------
static const size_t N_ACC     = (size_t)3 * HW * FEAT;   // 6,291,456
static const size_t OFF_CNT   = N_ACC;
static const size_t N_CNT     = (size_t)3 * HW;          //   196,608
static const size_t OFF_STATS = OFF_CNT + N_CNT;
static const size_t N_STATS   = 8;
static const size_t N_ZERO    = OFF_STATS + N_STATS;

// LDS byte offset of a __shared__ object: low 32 bits of its flat address
// (flat LDS addr = {SHARED_BASE, lds_offset} per ISA 10.2 aperture rules)
__device__ __forceinline__ unsigned lds_off(const void* p) {
  return (unsigned)(unsigned long long)p;
}

// async global->LDS copy of one dword per active lane (ASYNCcnt-tracked)
__device__ __forceinline__ void async_load_dword(unsigned lds_byte_off,
                                                 const float* gptr) {
  asm volatile("global_load_async_to_lds_b32 %0, %1, off"
               :: "v"(lds_byte_off), "v"((unsigned long long)gptr)
               : "memory");
}
__device__ __forceinline__ void wait_asynccnt0() {
  asm volatile("s_wait_asynccnt 0x0" ::: "memory");
}

// normalized 5-tap gaussian, sigma=1: band(i,n) = kc[i-n] if 0<=i-n<=4 else 0
__device__ __forceinline__ float bandw(int i, int n) {
  int d = i - n;
  float r = 0.0f;
  r = (d == 0 || d == 4) ? 0.054488684549642945f : r;
  r = (d == 1 || d == 3) ? 0.244201342003233160f : r;
  r = (d == 2)           ? 0.402619946894247790f : r;
  return r;
}

// ---------------- K1: zero scratch ----------------
__global__ void k_zero(float* __restrict__ ws, size_t n) {
  size_t i = (size_t)blockIdx.x * blockDim.x + threadIdx.x;
  size_t stride = (size_t)gridDim.x * blockDim.x;
  for (; i < n; i += stride) ws[i] = 0.0f;
}

// ---------------- K2: bilinear scatter (one wave32 per (gaussian, plane)) ----
__global__ void k_scatter(const float* __restrict__ feats,
                          const float* __restrict__ xyz,
                          float* __restrict__ acc,
                          float* __restrict__ counts) {
  long long gid = (long long)blockIdx.x * blockDim.x + threadIdx.x;
  long long wave = gid >> 5;
  int lane = (int)(gid & 31);
  if (wave >= (long long)NG * 3) return;
  int g = (int)(wave / 3);
  int p = (int)(wave - (long long)g * 3);

  float f = feats[(size_t)g * 96 + p * 32 + lane];   // coalesced 128B per wave
  float gx = xyz[(size_t)g * 3 + 0];
  float gy = xyz[(size_t)g * 3 + 1];
  float gz = xyz[(size_t)g * 3 + 2];

  float x = fminf(fmaxf((gx + 1.0f) * 0.5f, 0.0f), 0.999f) * (RES - 1);
  float y = fminf(fmaxf((gy + 1.0f) * 0.5f, 0.0f), 0.999f) * (RES - 1);
  float z = fminf(fmaxf((gz + 1.0f) * 0.5f, 0.0f), 0.999f) * (RES - 1);

  float u, v;            // u = idx[:,0] (px), v = idx[:,1] (py)
  if (p == 0)      { u = x; v = y; }   // plane_xy
  else if (p == 1) { u = x; v = z; }   // plane_xz
  else             { u = y; v = z; }   // plane_yz

  int fu = (int)floorf(u), fv = (int)floorf(v);
  int u0 = min(max(fu, 0), RES - 1);
  int u1 = min(max(fu + 1, 0), RES - 1);
  int v0 = min(max(fv, 0), RES - 1);
  int v1 = min(max(fv + 1, 0), RES - 1);
  float wu0 = fminf(fmaxf((float)u1 - u, 0.0f), 1.0f);
  float wu1 = fminf(fmaxf(u - (float)u0, 0.0f), 1.0f);
  float wv0 = fminf(fmaxf((float)v1 - v, 0.0f), 1.0f);
  float wv1 = fminf(fmaxf(v - (float)v0, 0.0f), 1.0f);

  int lin0 = v0 * RES + u0, lin1 = v1 * RES + u0;
  int lin2 = v0 * RES + u1, lin3 = v1 * RES + u1;
  float w0 = wu0 * wv0, w1 = wu0 * wv1, w2 = wu1 * wv0, w3 = wu1 * wv1;

  float* accp = acc + (size_t)p * PLANE_ELEMS;
  atomicAdd(&accp[(size_t)lin0 * FEAT + lane], f * w0);
  atomicAdd(&accp[(size_t)lin1 * FEAT + lane], f * w1);
  atomicAdd(&accp[(size_t)lin2 * FEAT + lane], f * w2);
  atomicAdd(&accp[(size_t)lin3 * FEAT + lane], f * w3);

  if (lane < 4) {
    int lin = (lane == 0) ? lin0 : (lane == 1) ? lin1 : (lane == 2) ? lin2 : lin3;
    atomicAdd(&counts[(size_t)p * HW + lin], 1.0f);
  }
}

// ---------------- K3: sum / sumsq reduction per plane ----------------
__global__ void k_reduce(const float* __restrict__ acc,
                         const float* __restrict__ counts,
                         float* __restrict__ stats) {
  int p = blockIdx.y;
  const float* accp = acc + (size_t)p * PLANE_ELEMS;
  const float* cntp = counts + (size_t)p * HW;

  float s = 0.0f, ss = 0.0f;
  for (size_t i = (size_t)blockIdx.x * blockDim.x + threadIdx.x;
       i < (size_t)PLANE_ELEMS;
       i += (size_t)gridDim.x * blockDim.x) {
    int pix = (int)(i >> 5);                     // acc layout [pix][feat]
    float v = accp[i] / (cntp[pix] + 1e-6f);
    s += v; ss += v * v;
  }
  for (int off = 16; off > 0; off >>= 1) {
    s  += __shfl_down(s,  off, 32);
    ss += __shfl_down(ss, off, 32);
  }
  __shared__ float shs[8], shss[8];
  int w = threadIdx.x >> 5, l = threadIdx.x & 31;
  if (l == 0) { shs[w] = s; shss[w] = ss; }
  __syncthreads();
  if (threadIdx.x == 0) {
    float S = 0.0f, SS = 0.0f;
    for (int i = 0; i < 8; i++) { S += shs[i]; SS += shss[i]; }
    atomicAdd(&stats[p * 2 + 0], S);
    atomicAdd(&stats[p * 2 + 1], SS);
  }
}

// ---------------- K4: normalize + LayerNorm + HWC->CHW transpose -------------
// 32x32 tile staged via async global->LDS scatter into padded layout
__global__ void k_normalize(const float* __restrict__ acc,
                            const float* __restrict__ counts,
                            const float* __restrict__ stats,
                            const float* __restrict__ lnw,
                            const float* __restrict__ lnb,
                            float* __restrict__ outNorm) {
  int p = blockIdx.y;
  int pixBase = blockIdx.x * 32;
  const float* accp = acc + (size_t)p * PLANE_ELEMS;

  float mu  = stats[p * 2 + 0] * (1.0f / PLANE_ELEMS);
  float var = stats[p * 2 + 1] * (1.0f / PLANE_ELEMS) - mu * mu;
  float inv = rsqrtf(var + 1e-5f);

  __shared__ float tile[32][33];   // padded: conflict-free transposed reads
  __shared__ float cnt[32];

  #pragma unroll
  for (int r = 0; r < 4; r++) {    // 1024 dwords, coalesced global, LDS scatter
    int idx = r * 256 + threadIdx.x;
    int pi = idx >> 5, c = idx & 31;
    async_load_dword(lds_off(&tile[pi][c]), accp + (size_t)pixBase * FEAT + idx);
  }
  if (threadIdx.x < 32) cnt[threadIdx.x] = counts[(size_t)p * HW + pixBase + threadIdx.x];
  wait_asynccnt0();
  __syncthreads();

  #pragma unroll
  for (int r = 0; r < 4; r++) {                   // transposed write, coalesced
    int idx = r * 256 + threadIdx.x;
    int c = idx >> 5, pi = idx & 31;
    float v = tile[pi][c] / (cnt[pi] + 1e-6f);
    size_t o = (size_t)c * HW + pixBase + pi;     // CHW offset
    outNorm[(size_t)p * PLANE_ELEMS + o] = (v - mu) * inv * lnw[o] + lnb[o];
  }
}

// ---------------- K5: separable 5-tap gaussian blur via banded WMMA ----------
// One wave32 per (plane, channel, 16x16 tile).
//  hblur: D = A(data 16x4 chunks) x B(band 4x16 chunks)  -> staged in LDS
//  vblur: D = A(band 16x4 chunks) x B(hblur 4x16 chunks)
__global__ void __launch_bounds__(32) k_blur(const float* __restrict__ norm,
                                             const float* __restrict__ pxy,
                                             const float* __restrict__ pxz,
                                             const float* __restrict__ pyz,
                                             float* __restrict__ outFinal) {
  int b   = blockIdx.x;          // 3 * 32 * 256 tiles
  int p   = b >> 13;             // / (32*256)
  int rem = b & 8191;
  int ch  = rem >> 8;
  int t   = rem & 255;
  int h0  = (t >> 4) * 16, w0 = (t & 15) * 16;

  const float* src   = norm + (size_t)p * PLANE_ELEMS + (size_t)ch * HW;
  const float* plane = ((p == 0) ? pxy : (p == 1) ? pxz : pyz) + (size_t)ch * HW;
  float* dst         = outFinal + (size_t)p * PLANE_ELEMS + (size_t)ch * HW;

  __shared__ float ext[20][20];   // input tile + 2-halo, zero padded (SAME)
  __shared__ float hb[20][16];    // horizontally blurred, rows = ext rows

  int lane = threadIdx.x;
  bool hi  = lane >= 16;
  int l15  = lane & 15;
  int k_lo = hi ? 2 : 0;
  int k_hi = hi ? 3 : 1;

  // stage halo tile: async global->LDS for in-bounds cells, DS-store zeros for
  // OOB cells (disjoint addresses, so no DS<->ASYNC ordering hazard)
  for (int idx = lane; idx < 400; idx += 32) {
    int er = idx / 20, ec = idx - er * 20;
    int h = h0 - 2 + er, w = w0 - 2 + ec;
    if (h >= 0 && h < RES && w >= 0 && w < RES) {
      async_load_dword(lds_off(&ext[er][ec]), src + h * RES + w);
    } else {
      ext[er][ec] = 0.0f;
    }
  }
  wait_asynccnt0();
  __syncthreads();

  // band chunks, shared between hblur-B and vblur-A
  v2f bw[5];
  #pragma unroll
  for (int c = 0; c < 5; c++) {
    bw[c].x = bandw(4 * c + k_lo, l15);
    bw[c].y = bandw(4 * c + k_hi, l15);
  }

  // horizontal pass: two row-shifted tiles cover ext rows 0..15 and 4..19
  v8f d0 = {};
  v8f d1 = {};
  #pragma unroll
  for (int c = 0; c < 5; c++) {
    int k0 = 4 * c;
    v2f a0, a1;
    a0.x = ext[l15 + 0][k0 + k_lo];  a0.y = ext[l15 + 0][k0 + k_hi];
    a1.x = ext[l15 + 4][k0 + k_lo];  a1.y = ext[l15 + 4][k0 + k_hi];
    d0 = __builtin_amdgcn_wmma_f32_16x16x4_f32(false, a0, false, bw[c],
                                               (short)0, d0, false, false);
    d1 = __builtin_amdgcn_wmma_f32_16x16x4_f32(false, a1, false, bw[c],
                                               (short)0, d1, false, false);
  }
  int rowbase = hi ? 8 : 0;
  #pragma unroll
  for (int v = 0; v < 8; v++) hb[v + rowbase][l15] = d0[v];   // ext rows 0..15
  if (hi) {
    #pragma unroll
    for (int v = 4; v < 8; v++) hb[v + 12][l15] = d1[v];      // ext rows 16..19
  }
  __syncthreads();

  // vertical pass: A = band, B = hb rows
  v8f dv = {};
  #pragma unroll
  for (int c = 0; c < 5; c++) {
    int k0 = 4 * c;
    v2f bb;
    bb.x = hb[k0 + k_lo][l15];
    bb.y = hb[k0 + k_hi][l15];
    dv = __builtin_amdgcn_wmma_f32_16x16x4_f32(false, bw[c], false, bb,
                                               (short)0, dv, false, false);
  }

  #pragma unroll
  for (int v = 0; v < 8; v++) {
    int h = h0 + v + rowbase, w = w0 + l15;
    dst[h * RES + w] = dv[v] + plane[h * RES + w];
  }
}

// ---------------- host launch ----------------
extern "C" void kernel_launch(void* const* d_in, const int* in_sizes, int n_in,
                              void* d_out, int out_size, void* d_ws, size_t ws_size,
                              hipStream_t stream) {
  const float* plane_xy = (const float*)d_in[0];
  const float* plane_xz = (const float*)d_in[1];
  const float* plane_yz = (const float*)d_in[2];
  const float* ln_w     = (const float*)d_in[3];
  const float* ln_b     = (const float*)d_in[4];
  const float* feats    = (const float*)d_in[5];
  const float* xyz      = (const float*)d_in[6];
  float* out = (float*)d_out;
  float* ws  = (float*)d_ws;

  float* acc    = ws;                 // [3][HW][FEAT], later reused for blur output
  float* counts = ws + OFF_CNT;       // [3][HW]
  float* stats  = ws + OFF_STATS;     // [3][2]

  // K1: zero acc / counts / stats
  k_zero<<<8192, 256, 0, stream>>>(ws, N_ZERO);

  // K2: scatter (3 * 500k waves)
  {
    long long threads = (long long)3 * NG * 32;
    int blocks = (int)((threads + 255) / 256);
    k_scatter<<<blocks, 256, 0, stream>>>(feats, xyz, acc, counts);
  }

  // K3: per-plane mean/var moments
  k_reduce<<<dim3(1024, 3), 256, 0, stream>>>(acc, counts, stats);

  // K4: normalize + layernorm, CHW layout into d_out (temp)
  k_normalize<<<dim3(HW / 32, 3), 256, 0, stream>>>(acc, counts, stats,
                                                    ln_w, ln_b, out);

  // K5: WMMA separable blur + plane add, into ws acc region (now dead)
  k_blur<<<3 * FEAT * 256, 32, 0, stream>>>(out, plane_xy, plane_xz, plane_yz, acc);

  // final result -> d_out
  hipMemcpyAsync(out, acc, (size_t)3 * PLANE_ELEMS * sizeof(float),
                 hipMemcpyDeviceToDevice, stream);
}